// RecurrentInhibition_6030134083806
// MI455X (gfx1250) — compile-verified
//
#include <hip/hip_runtime.h>

typedef float v2f __attribute__((ext_vector_type(2)));
typedef float v8f __attribute__((ext_vector_type(8)));

#define SCOPE       27
#define HALF_SCOPE  13
#define DECAY       0.05f
#define MAX_STEPS   10
#define NCH         128
#define HW          3136          // 56*56
#define NBATCH      32
#define TILE_N      16
#define TILES_PB    (HW / TILE_N) // 196

// ---------------------------------------------------------------------------
// Kernel 1: build the effective matrix E = P_10, where
//   P_0 = I,  P_t = M * P_{t-1} + DECAY*I,  M = (1-DECAY)*I + DECAY*W,
// W = banded Toeplitz from w_rec (cross-correlation along channels, zero pad).
// Columns of P are independent: one block per column j.
// Output layout in d_ws: WMMA f32 16x16x4 A-fragments, per (m-block, k-block):
//   float2 frag[(m*32+kb)*32 + lane] = { A_vgpr0, A_vgpr1 }
//   lane<16 : { E[16m+lane][4kb+0], E[16m+lane][4kb+1] }
//   lane>=16: { E[16m+lane-16][4kb+2], E[16m+lane-16][4kb+3] }
// ---------------------------------------------------------------------------
__global__ void build_emat(const float* __restrict__ w_rec,
                           float* __restrict__ ws) {
  __shared__ float wsh[SCOPE];
  __shared__ float buf[2][NCH];
  const int i = threadIdx.x;  // row index 0..127
  const int j = blockIdx.x;   // column index 0..127
  if (i < SCOPE) wsh[i] = w_rec[i];
  buf[0][i] = (i == j) ? 1.0f : 0.0f;
  __syncthreads();
  int cur = 0;
  for (int t = 0; t < MAX_STEPS; ++t) {
    float s = 0.0f;
#pragma unroll
    for (int d = -HALF_SCOPE; d <= HALF_SCOPE; ++d) {
      const int k = i + d;
      if (k >= 0 && k < NCH) s += wsh[d + HALF_SCOPE] * buf[cur][k];
    }
    const float v = (1.0f - DECAY) * buf[cur][i] + DECAY * s +
                    ((i == j) ? DECAY : 0.0f);
    buf[cur ^ 1][i] = v;
    __syncthreads();
    cur ^= 1;
  }
  // Scatter E[i][j] into A-fragment layout.
  const int m    = i >> 4;
  const int kb   = j >> 2;
  const int ko   = j & 3;
  const int lane = (i & 15) + ((ko >> 1) << 4);
  const int slot = ko & 1;
  ws[(((m * 32 + kb) * 32 + lane) << 1) + slot] = buf[cur][i];
}

// ---------------------------------------------------------------------------
// Kernel 2: out[b, :, :] = E (128x128) x act[b, :, :] (128 x 3136), done as
// one 128x16 output tile per wave using v_wmma_f32_16x16x4_f32.
// B-fragment layout (4x16, K x N), mirroring the documented A/CD half-split:
//   v0: lanes 0-15 = row K=0 (N=lane), lanes 16-31 = row K=2 (N=lane-16)
//   v1: lanes 0-15 = row K=1,          lanes 16-31 = row K=3
// D layout (ISA): VGPR g -> rows g (lanes 0-15) and g+8 (lanes 16-31).
// ---------------------------------------------------------------------------
__global__ __launch_bounds__(256)
void apply_emat(const float* __restrict__ act,
                const float* __restrict__ ws,
                float* __restrict__ out) {
  const int lane = threadIdx.x & 31;
  const int wave = (blockIdx.x << 3) + (threadIdx.x >> 5);
  const int b    = wave / TILES_PB;
  const int t    = wave - b * TILES_PB;
  const int p0   = t * TILE_N;
  const int half = lane >> 4;
  const int li   = lane & 15;

  // Load the whole 128 x 16 B-tile into registers as 32 K-fragments.
  const float* xb = act + (size_t)b * NCH * HW + p0 + li;
  v2f bf[32];
#pragma unroll
  for (int kb = 0; kb < 32; ++kb) {
    const float* p = xb + (size_t)(4 * kb + 2 * half) * HW;
    v2f v;
    v.x = p[0];    // K = 4kb + 2*half
    v.y = p[HW];   // K = 4kb + 2*half + 1
    bf[kb] = v;
  }

  const v2f* frag = ((const v2f*)ws) + lane;
  float* ob = out + (size_t)b * NCH * HW + p0 + li;

#pragma unroll
  for (int m = 0; m < 8; ++m) {
    v8f acc = {0.f, 0.f, 0.f, 0.f, 0.f, 0.f, 0.f, 0.f};
    const v2f* fm = frag + (size_t)m * 32 * 32;
#pragma unroll
    for (int kb = 0; kb < 32; ++kb) {
      const v2f af = fm[kb << 5];
      acc = __builtin_amdgcn_wmma_f32_16x16x4_f32(
          false, af, false, bf[kb], (short)0, acc, false, false);
    }
    float* op = ob + (size_t)(16 * m + 8 * half) * HW;
#pragma unroll
    for (int g = 0; g < 8; ++g) op[(size_t)g * HW] = acc[g];
  }
}

// ---------------------------------------------------------------------------
extern "C" void kernel_launch(void* const* d_in, const int* in_sizes, int n_in,
                              void* d_out, int out_size, void* d_ws,
                              size_t ws_size, hipStream_t stream) {
  (void)in_sizes; (void)n_in; (void)out_size; (void)ws_size;
  const float* act   = (const float*)d_in[0];
  const float* w_rec = (const float*)d_in[1];
  float* out = (float*)d_out;
  float* ws  = (float*)d_ws;  // needs 128*128*4 = 64 KB

  build_emat<<<NCH, NCH, 0, stream>>>(w_rec, ws);

  const int total_waves = NBATCH * TILES_PB;      // 6272
  const int blocks      = total_waves / 8;        // 784 blocks x 8 waves
  apply_emat<<<blocks, 256, 0, stream>>>(act, ws, out);
}